// ActSeries_48292612276615
// MI455X (gfx1250) — compile-verified
//
#include <hip/hip_runtime.h>

// ============================================================================
// ActSeries: 20 x (per-channel range-norm + quadratic) on [32,32,256,256] f32.
// Memory-bound: ~10.5 GB total traffic -> ~0.45 ms at 23.3 TB/s.
// CDNA5 path: global_load_async_to_lds_b128 streaming + s_wait_asynccnt,
// deterministic per-channel min/max via monotonic uint keys + atomicMin/Max.
// ============================================================================

#define THREADS   256
#define NWAVES    8          // 256 / wave32
#define CHUNK_F   128        // floats per wave-chunk: 32 lanes * 4 floats
#define NCHUNK    16         // chunks per wave  -> 2048 floats per wave
#define DEPTH     4          // async pipeline depth (LDS slots per wave)
#define BLOCK_F   16384      // floats per block = NWAVES*NCHUNK*CHUNK_F
#define NCH       32
#define NLAYERS   20
#define TOTAL_F   (32 * NCH * 256 * 256)   // 2^26
#define EPSV      1e-5f

#if defined(__HIP_DEVICE_COMPILE__) && __has_builtin(__builtin_amdgcn_global_load_async_to_lds_b128)
#define USE_ASYNC_BUILTIN 1
#else
#define USE_ASYNC_BUILTIN 0
#endif

typedef int v4i_ __attribute__((ext_vector_type(4)));
typedef __attribute__((address_space(1))) v4i_* gptr_v4i;
typedef __attribute__((address_space(3))) v4i_* lptr_v4i;

// ---------------- monotonic float <-> uint keys (order-preserving) ----------
__device__ __forceinline__ unsigned fkey(float f) {
  unsigned u = __float_as_uint(f);
  return u ^ (unsigned)(((int)u >> 31) | (int)0x80000000);
}
__device__ __forceinline__ float funkey(unsigned k) {
  unsigned m = (unsigned)(((int)(~k) >> 31) | (int)0x80000000);
  return __uint_as_float(k ^ m);
}
__device__ __forceinline__ unsigned umn(unsigned a, unsigned b) { return a < b ? a : b; }
__device__ __forceinline__ unsigned umx(unsigned a, unsigned b) { return a > b ? a : b; }

// ---------------- CDNA5 async global->LDS copy (16B per lane) ---------------
__device__ __forceinline__ void asyncCopy16(const float* g, float* l) {
#if USE_ASYNC_BUILTIN
  __builtin_amdgcn_global_load_async_to_lds_b128(
      (gptr_v4i)(void*)g,
      (lptr_v4i)(void*)l,
      0, 0);
#elif defined(__HIP_DEVICE_COMPILE__)
  // Flat LDS addresses: low 32 bits are the LDS byte offset (ISA 10.2).
  asm volatile("global_load_async_to_lds_b128 %0, %1, off"
               :: "v"((unsigned)(unsigned long long)(void*)l),
                  "v"((unsigned long long)(const void*)g)
               : "memory");
#else
  (void)g; (void)l;
#endif
}

template <int N>
__device__ __forceinline__ void waitAsync() {
#if defined(__HIP_DEVICE_COMPILE__) && __has_builtin(__builtin_amdgcn_s_wait_asynccnt)
  __builtin_amdgcn_s_wait_asynccnt(N);
  asm volatile("" ::: "memory");
#elif defined(__HIP_DEVICE_COMPILE__)
  asm volatile("s_wait_asynccnt %0" :: "i"(N) : "memory");
#endif
}

// ---------------- stats init: min-keys = 0xFFFFFFFF, max-keys = 0 -----------
__global__ void ActSeries_init_stats(unsigned* __restrict__ stats) {
  int i = blockIdx.x * blockDim.x + threadIdx.x;
  if (i < (NLAYERS + 1) * 2 * NCH)
    stats[i] = ((i & 63) < NCH) ? 0xFFFFFFFFu : 0u;
}

// ---------------- stage-0 reduction over x ----------------------------------
__global__ __launch_bounds__(THREADS)
void ActSeries_reduce(const float* __restrict__ in, unsigned* __restrict__ stats) {
  __shared__ unsigned sMin, sMax;
  const int tid = threadIdx.x;
  const int base = blockIdx.x * BLOCK_F;
  const int c = (base >> 16) & (NCH - 1);   // plane = base / 65536, channel = plane % 32

  if (tid == 0) { sMin = 0xFFFFFFFFu; sMax = 0u; }
  __syncthreads();

  const float4* in4 = (const float4*)(in + base);
  unsigned kmin = 0xFFFFFFFFu, kmax = 0u;
#pragma unroll
  for (int i = 0; i < 16; ++i) {
    float4 v = in4[tid + i * THREADS];
    unsigned k0 = fkey(v.x), k1 = fkey(v.y), k2 = fkey(v.z), k3 = fkey(v.w);
    kmin = umn(umn(umn(kmin, k0), k1), umn(k2, k3));
    kmax = umx(umx(umx(kmax, k0), k1), umx(k2, k3));
  }
#pragma unroll
  for (int off = 16; off > 0; off >>= 1) {
    kmin = umn(kmin, (unsigned)__shfl_xor((int)kmin, off, 32));
    kmax = umx(kmax, (unsigned)__shfl_xor((int)kmax, off, 32));
  }
  if ((tid & 31) == 0) { atomicMin(&sMin, kmin); atomicMax(&sMax, kmax); }
  __syncthreads();
  if (tid == 0) {
    atomicMin(&stats[c], sMin);
    atomicMax(&stats[NCH + c], sMax);
  }
}

// ---------------- fused layer: normalize+poly, accumulate next stats --------
__global__ __launch_bounds__(THREADS)
void ActSeries_layer(const float* __restrict__ in, float* __restrict__ out,
                     const unsigned* __restrict__ statsIn, unsigned* __restrict__ statsOut,
                     const float* __restrict__ w0, const float* __restrict__ w1,
                     const float* __restrict__ w2, int layer, int updateNext) {
  __shared__ __align__(16) float tile[NWAVES * DEPTH * CHUNK_F];  // 16 KB
  __shared__ unsigned sMin, sMax;

  const int tid  = threadIdx.x;
  const int wave = tid >> 5;
  const int lane = tid & 31;
  const int base = blockIdx.x * BLOCK_F;
  const int c = (base >> 16) & (NCH - 1);   // block-uniform channel

  const float mn = funkey(statsIn[c]);
  const float mx = funkey(statsIn[NCH + c]);
  const float scale = 1.0f / (mx - mn + EPSV);
  const float a0 = w0[layer * NCH + c];
  const float a1 = w1[layer * NCH + c];
  const float a2 = w2[layer * NCH + c];

  if (tid == 0) { sMin = 0xFFFFFFFFu; sMax = 0u; }
  __syncthreads();

  const float* gin  = in  + base + wave * (NCHUNK * CHUNK_F) + lane * 4;
  float*       gout = out + base + wave * (NCHUNK * CHUNK_F) + lane * 4;
  float*       lb   = &tile[wave * DEPTH * CHUNK_F + lane * 4];

  // prologue: keep DEPTH-1 = 3 chunks in flight per wave
  asyncCopy16(gin + 0 * CHUNK_F, lb + 0 * CHUNK_F);
  asyncCopy16(gin + 1 * CHUNK_F, lb + 1 * CHUNK_F);
  asyncCopy16(gin + 2 * CHUNK_F, lb + 2 * CHUNK_F);

  unsigned kmin = 0xFFFFFFFFu, kmax = 0u;

  for (int ci = 0; ci < NCHUNK; ++ci) {
    if (ci + 3 < NCHUNK) {
      asyncCopy16(gin + (ci + 3) * CHUNK_F, lb + ((ci + 3) & (DEPTH - 1)) * CHUNK_F);
      waitAsync<3>();                       // chunk ci retired
    } else if (ci == NCHUNK - 3) waitAsync<2>();
    else if (ci == NCHUNK - 2)   waitAsync<1>();
    else                         waitAsync<0>();

    const float4 v = *(const float4*)(lb + (ci & (DEPTH - 1)) * CHUNK_F);
    float4 y;
    {
      float xh = (v.x - mn) * scale; y.x = __builtin_fmaf(__builtin_fmaf(a2, xh, a1), xh, a0);
      xh = (v.y - mn) * scale;       y.y = __builtin_fmaf(__builtin_fmaf(a2, xh, a1), xh, a0);
      xh = (v.z - mn) * scale;       y.z = __builtin_fmaf(__builtin_fmaf(a2, xh, a1), xh, a0);
      xh = (v.w - mn) * scale;       y.w = __builtin_fmaf(__builtin_fmaf(a2, xh, a1), xh, a0);
    }
    *(float4*)(gout + ci * CHUNK_F) = y;

    unsigned k0 = fkey(y.x), k1 = fkey(y.y), k2 = fkey(y.z), k3 = fkey(y.w);
    kmin = umn(umn(umn(kmin, k0), k1), umn(k2, k3));
    kmax = umx(umx(umx(kmax, k0), k1), umx(k2, k3));
  }

  if (updateNext) {
#pragma unroll
    for (int off = 16; off > 0; off >>= 1) {
      kmin = umn(kmin, (unsigned)__shfl_xor((int)kmin, off, 32));
      kmax = umx(kmax, (unsigned)__shfl_xor((int)kmax, off, 32));
    }
    if (lane == 0) { atomicMin(&sMin, kmin); atomicMax(&sMax, kmax); }
    __syncthreads();
    if (tid == 0) {
      atomicMin(&statsOut[c], sMin);
      atomicMax(&statsOut[NCH + c], sMax);
    }
  }
}

// ---------------- host entry ------------------------------------------------
extern "C" void kernel_launch(void* const* d_in, const int* in_sizes, int n_in,
                              void* d_out, int out_size, void* d_ws, size_t ws_size,
                              hipStream_t stream) {
  (void)in_sizes; (void)n_in; (void)out_size; (void)ws_size;
  const float* x  = (const float*)d_in[0];
  const float* w0 = (const float*)d_in[1];
  const float* w1 = (const float*)d_in[2];
  const float* w2 = (const float*)d_in[3];
  float* out = (float*)d_out;
  unsigned* stats = (unsigned*)d_ws;        // (NLAYERS+1) * 64 u32 = 5376 B

  const int nBlocks = TOTAL_F / BLOCK_F;    // 4096

  ActSeries_init_stats<<<6, 256, 0, stream>>>(stats);
  ActSeries_reduce<<<nBlocks, THREADS, 0, stream>>>(x, stats);
  for (int l = 0; l < NLAYERS; ++l) {
    const float* src = (l == 0) ? x : out;  // layers 2..20 run in place on out
    ActSeries_layer<<<nBlocks, THREADS, 0, stream>>>(
        src, out, stats + l * 64, stats + (l + 1) * 64,
        w0, w1, w2, l, (l < NLAYERS - 1) ? 1 : 0);
  }
}